// GINConvLayer_80711025426653
// MI455X (gfx1250) — compile-verified
//
#include <hip/hip_runtime.h>

#define N_NODES 100000
#define N_EDGES 1600000
#define DIM 64

typedef float v2f __attribute__((ext_vector_type(2)));
typedef float v8f __attribute__((ext_vector_type(8)));

// ---------------- Kernel 1: hbuf = x ----------------
__global__ void gin_init_kernel(const float4* __restrict__ x4,
                                float4* __restrict__ h4, int n4) {
    int i = blockIdx.x * blockDim.x + threadIdx.x;
    if (i < n4) h4[i] = x4[i];
}

// ---------------- Kernel 2: scatter-add x[src] into hbuf[dst] ----------------
// 16 lanes per edge, each lane: one float4 load + 4 f32 atomic adds.
__global__ void gin_scatter_kernel(const float* __restrict__ x,
                                   const int* __restrict__ ei,
                                   float* __restrict__ hbuf) {
    int tid = blockIdx.x * blockDim.x + threadIdx.x;
    int e = tid >> 4;
    if (e >= N_EDGES) return;
    int j = (tid & 15) * 4;
    int src = ei[e];
    int dst = ei[N_EDGES + e];
    const float4 v = *(const float4*)(x + (size_t)src * DIM + j);
    float* p = hbuf + (size_t)dst * DIM + j;
    unsafeAtomicAdd(p + 0, v.x);
    unsafeAtomicAdd(p + 1, v.y);
    unsafeAtomicAdd(p + 2, v.z);
    unsafeAtomicAdd(p + 3, v.w);
}

// ---------------- Kernel 3: MLP + residual, f32 WMMA ----------------
// block = 128 threads = 4 waves; each wave owns a 16-node M-tile (64 nodes/block).
__global__ __launch_bounds__(128) void gin_mlp_kernel(
    const float* __restrict__ x, const float* __restrict__ hbuf,
    const float* __restrict__ W1, const float* __restrict__ b1,
    const float* __restrict__ W2, const float* __restrict__ b2,
    float* __restrict__ out) {
    __shared__ float sW1[DIM * 65];   // padded stride 65
    __shared__ float sW2[DIM * 65];
    __shared__ float sH[64 * 68];     // padded stride 68 (conflict-free A-frag reads)
    __shared__ float sT[64 * 68];
    __shared__ float sb1[DIM];
    __shared__ float sb2[DIM];

    const int t = threadIdx.x;
    const int blockStart = blockIdx.x * 64;

    // Stage weights + biases (coalesced).
    for (int i = 0; i < 32; ++i) {
        int idx = i * 128 + t;            // 0..4095
        int k = idx >> 6, n = idx & 63;
        sW1[k * 65 + n] = W1[idx];
        sW2[k * 65 + n] = W2[idx];
    }
    if (t < DIM) { sb1[t] = b1[t]; sb2[t] = b2[t]; }

    // Stage 64x64 h tile; each thread loads a half-row as float4s.
    {
        int g = t >> 1, half = t & 1;
        int node = blockStart + g;
        float* dstp = &sH[g * 68 + half * 32];
        if (node < N_NODES) {
            const float4* srcp = (const float4*)(hbuf + (size_t)node * DIM + half * 32);
            #pragma unroll
            for (int i = 0; i < 8; ++i) {
                float4 v = srcp[i];
                dstp[4 * i + 0] = v.x; dstp[4 * i + 1] = v.y;
                dstp[4 * i + 2] = v.z; dstp[4 * i + 3] = v.w;
            }
        } else {
            #pragma unroll
            for (int i = 0; i < 32; ++i) dstp[i] = 0.0f;
        }
    }
    __syncthreads();

    const int lane  = t & 31;
    const int wv    = t >> 5;
    const int gbase = wv * 16;          // wave-local M-tile base row
    const int m     = lane & 15;
    const int hi    = lane >> 4;        // half-wave selector (K pairs / M+8 rows)

    // GEMM1: sT = relu(sH @ W1 + b1)
    for (int nt = 0; nt < 4; ++nt) {
        const int n = m + nt * 16;
        v8f acc = {0.f, 0.f, 0.f, 0.f, 0.f, 0.f, 0.f, 0.f};
        #pragma unroll
        for (int kk = 0; kk < 16; ++kk) {
            const int kb = kk * 4 + hi * 2;
            v2f a; a.x = sH[(gbase + m) * 68 + kb];
                   a.y = sH[(gbase + m) * 68 + kb + 1];
            v2f b; b.x = sW1[kb * 65 + n];
                   b.y = sW1[(kb + 1) * 65 + n];
            acc = __builtin_amdgcn_wmma_f32_16x16x4_f32(
                false, a, false, b, (short)0, acc, false, false);
        }
        const float bias = sb1[n];
        #pragma unroll
        for (int r = 0; r < 8; ++r) {
            float v = acc[r] + bias;
            sT[(gbase + r + 8 * hi) * 68 + n] = v > 0.f ? v : 0.f;
        }
    }
    // No barrier needed: each wave reads back only the sT rows it wrote.

    // GEMM2: out = x + relu(sT @ W2 + b2)
    for (int nt = 0; nt < 4; ++nt) {
        const int n = m + nt * 16;
        v8f acc = {0.f, 0.f, 0.f, 0.f, 0.f, 0.f, 0.f, 0.f};
        #pragma unroll
        for (int kk = 0; kk < 16; ++kk) {
            const int kb = kk * 4 + hi * 2;
            v2f a; a.x = sT[(gbase + m) * 68 + kb];
                   a.y = sT[(gbase + m) * 68 + kb + 1];
            v2f b; b.x = sW2[kb * 65 + n];
                   b.y = sW2[(kb + 1) * 65 + n];
            acc = __builtin_amdgcn_wmma_f32_16x16x4_f32(
                false, a, false, b, (short)0, acc, false, false);
        }
        const float bias = sb2[n];
        #pragma unroll
        for (int r = 0; r < 8; ++r) {
            const int node = blockStart + gbase + r + 8 * hi;
            if (node < N_NODES) {
                float v = acc[r] + bias;
                v = v > 0.f ? v : 0.f;
                out[(size_t)node * DIM + n] = x[(size_t)node * DIM + n] + v;
            }
        }
    }
}

extern "C" void kernel_launch(void* const* d_in, const int* in_sizes, int n_in,
                              void* d_out, int out_size, void* d_ws, size_t ws_size,
                              hipStream_t stream) {
    const float* x  = (const float*)d_in[0];
    const int*   ei = (const int*)d_in[1];
    const float* W1 = (const float*)d_in[2];
    const float* b1 = (const float*)d_in[3];
    const float* W2 = (const float*)d_in[4];
    const float* b2 = (const float*)d_in[5];
    float* out  = (float*)d_out;
    float* hbuf = (float*)d_ws;   // N_NODES*DIM floats = 25.6 MB

    // 1) hbuf = x
    {
        int n4 = N_NODES * DIM / 4;   // 1,600,000 float4
        int blk = 256, grd = (n4 + blk - 1) / blk;
        gin_init_kernel<<<grd, blk, 0, stream>>>((const float4*)x, (float4*)hbuf, n4);
    }
    // 2) scatter-add over edges
    {
        long long total = (long long)N_EDGES * 16;
        int blk = 256;
        int grd = (int)((total + blk - 1) / blk);
        gin_scatter_kernel<<<grd, blk, 0, stream>>>(x, ei, hbuf);
    }
    // 3) WMMA MLP + residual
    {
        int grd = (N_NODES + 63) / 64;  // 1563
        gin_mlp_kernel<<<grd, 128, 0, stream>>>(x, hbuf, W1, b1, W2, b2, out);
    }
}